// GNNGuard_GCN_60799557042764
// MI455X (gfx1250) — compile-verified
//
#include <hip/hip_runtime.h>
#include <math.h>

typedef __attribute__((ext_vector_type(2))) float v2f;
typedef __attribute__((ext_vector_type(8))) float v8f;

#define FEAT 128

__device__ __forceinline__ float wave_sum(float v) {
#pragma unroll
  for (int off = 16; off > 0; off >>= 1) v += __shfl_xor(v, off, 32);
  return v;
}
__device__ __forceinline__ float wave_max(float v) {
#pragma unroll
  for (int off = 16; off > 0; off >>= 1) v = fmaxf(v, __shfl_xor(v, off, 32));
  return v;
}

// ---------------- attention (edge weight) kernels ----------------

// one wave32 per node: L2-normalize its 128 features (float4 per lane)
__global__ void node_l2norm(const float* __restrict__ x, float* __restrict__ xn, int n) {
  int node = blockIdx.x * 8 + (threadIdx.x >> 5);
  int lane = threadIdx.x & 31;
  if (node >= n) return;
  const float4* px = reinterpret_cast<const float4*>(x + (size_t)node * FEAT);
  float4 v = px[lane];
  float s = v.x * v.x + v.y * v.y + v.z * v.z + v.w * v.w;
  s = wave_sum(s);
  float inv = 1.0f / fmaxf(sqrtf(s), 1e-12f);
  float4 r;
  r.x = v.x * inv; r.y = v.y * inv; r.z = v.z * inv; r.w = v.w * inv;
  reinterpret_cast<float4*>(xn + (size_t)node * FEAT)[lane] = r;
}

// one wave32 per edge: cosine similarity, threshold, zero diagonal
__global__ void edge_cos_sim(const float* __restrict__ xn, const int* __restrict__ row,
                             const int* __restrict__ col, float* __restrict__ sim, int E) {
  int e = blockIdx.x * 8 + (threadIdx.x >> 5);
  int lane = threadIdx.x & 31;
  if (e >= E) return;
  int r = row[e], c = col[e];
  const float4* pr = reinterpret_cast<const float4*>(xn + (size_t)r * FEAT);
  const float4* pc = reinterpret_cast<const float4*>(xn + (size_t)c * FEAT);
  float4 a = pr[lane], b = pc[lane];
  float s = a.x * b.x + a.y * b.y + a.z * b.z + a.w * b.w;
  s = wave_sum(s);
  if (lane == 0) {
    if (s < 0.1f) s = 0.0f;
    if (r == c) s = 0.0f;
    sim[e] = s;
  }
}

__global__ void edge_rowsum(const int* __restrict__ row, const float* __restrict__ sim,
                            float* __restrict__ rowsum, int E) {
  int e = blockIdx.x * blockDim.x + threadIdx.x;
  if (e >= E) return;
  float s = sim[e];
  if (s != 0.0f) atomicAdd(&rowsum[row[e]], s);
}

// row-L1 normalize in place; count nonzero entries per row (degree)
__global__ void edge_norm_deg(const int* __restrict__ row, float* __restrict__ sim,
                              const float* __restrict__ rowsum, float* __restrict__ deg, int E) {
  int e = blockIdx.x * blockDim.x + threadIdx.x;
  if (e >= E) return;
  int r = row[e];
  float rs = rowsum[r];
  float sn = (rs > 0.0f) ? sim[e] / rs : 0.0f;
  sim[e] = sn;
  if (sn != 0.0f) atomicAdd(&deg[r], 1.0f);
}

// self-loop gating, exp, optional gate-blend with base adjacency
__global__ void edge_weight_final(const int* __restrict__ row, const int* __restrict__ col,
                                  const float* __restrict__ sim, const float* __restrict__ deg,
                                  const float* __restrict__ adjbase, const float* __restrict__ gate,
                                  int gidx, float* __restrict__ wout, int E) {
  int e = blockIdx.x * blockDim.x + threadIdx.x;
  if (e >= E) return;
  int r = row[e], c = col[e];
  float w = (r == c) ? 1.0f / (deg[r] + 1.0f) : sim[e];
  float a = expf(w);
  if (gidx >= 0) {
    float g = gate[gidx];
    a = g * adjbase[e] + (1.0f - g) * a;
  }
  wout[e] = a;
}

// ---------------- GCN conv kernels ----------------

// H[n x nOut] = X[n x 128] @ W[128 x nOut], fp32 via V_WMMA_F32_16X16X4_F32.
// W is staged in LDS transposed (Wt[n][k]) and zero-padded to a multiple of 16
// columns, so the K-loop has no guards: one global_load_b64 (A pair) and one
// ds_load_b64 (B pair) per step, fully unrolled -> 32 back-to-back WMMAs.
// Layouts per ISA 7.12.2: lanes 0-15 carry K={0,1} (M/N = lane), lanes 16-31
// carry K={2,3}; D VGPR v holds row v (lanes 0-15) / row v+8 (lanes 16-31).
__global__ void gemm_wmma_f32(const float* __restrict__ X, const float* __restrict__ W,
                              float* __restrict__ H, int n, int nOut, int tilesN) {
  __shared__ float Wt[128 * FEAT];  // [col][k], 64KB max
  const int P = tilesN * 16;        // padded column count
  for (int idx = threadIdx.x; idx < P * FEAT; idx += blockDim.x) {
    int cn = idx >> 7;   // column
    int k  = idx & 127;  // k index
    Wt[idx] = (cn < nOut) ? W[(size_t)k * nOut + cn] : 0.0f;
  }
  __syncthreads();

  int wid = threadIdx.x >> 5;
  int tilesM = (n + 15) >> 4;
  int tile = blockIdx.x * 8 + wid;
  if (tile >= tilesM * tilesN) return;
  int tm = tile / tilesN, tn = tile % tilesN;
  int lane = threadIdx.x & 31;
  int half = lane >> 4;
  int lm = lane & 15;
  int m = tm * 16 + lm;
  if (m >= n) m = n - 1;            // clamp (never hit: N % 16 == 0)
  int nn = tn * 16 + lm;            // padded column owned by this lane

  const float* ap = X + (size_t)m * FEAT + 2 * half;   // 8B aligned
  const float* bp = Wt + (size_t)nn * FEAT + 2 * half; // 8B aligned
  v8f acc = {};
#pragma unroll
  for (int k = 0; k < FEAT; k += 4) {
    v2f a = *(const v2f*)(ap + k);
    v2f b = *(const v2f*)(bp + k);
    acc = __builtin_amdgcn_wmma_f32_16x16x4_f32(false, a, false, b,
                                                (short)0, acc, false, false);
  }
  if (nn < nOut) {
#pragma unroll
    for (int v = 0; v < 8; ++v) {
      int mm = tm * 16 + v + 8 * half;
      if (mm < n) H[(size_t)mm * nOut + nn] = acc[v];
    }
  }
}

__global__ void edge_colsum(const int* __restrict__ col, const float* __restrict__ w,
                            float* __restrict__ deg2, int E) {
  int e = blockIdx.x * blockDim.x + threadIdx.x;
  if (e >= E) return;
  atomicAdd(&deg2[col[e]], w[e]);
}

__global__ void node_rsqrt(const float* __restrict__ d, float* __restrict__ dinv, int n) {
  int i = blockIdx.x * blockDim.x + threadIdx.x;
  if (i >= n) return;
  float v = d[i];
  dinv[i] = (v > 0.0f) ? rsqrtf(v) : 0.0f;
}

// one wave32 per edge: out[col] += dinv[row]*w*dinv[col] * hlin[row]
__global__ void edge_aggregate(const int* __restrict__ row, const int* __restrict__ col,
                               const float* __restrict__ ew, const float* __restrict__ dinv,
                               const float* __restrict__ hlin, float* __restrict__ hout,
                               int E, int nOut) {
  int e = blockIdx.x * 8 + (threadIdx.x >> 5);
  int lane = threadIdx.x & 31;
  if (e >= E) return;
  int r = row[e], c = col[e];
  float nrm = dinv[r] * ew[e] * dinv[c];
  const float* hs = hlin + (size_t)r * nOut;
  float* hd = hout + (size_t)c * nOut;
  for (int j = lane; j < nOut; j += 32) atomicAdd(&hd[j], nrm * hs[j]);
}

__global__ void bias_act(float* __restrict__ h, const float* __restrict__ b,
                         long total, int nOut, int doRelu) {
  long i = (long)blockIdx.x * blockDim.x + threadIdx.x;
  if (i >= total) return;
  float v = h[i] + b[(int)(i % nOut)];
  if (doRelu) v = fmaxf(v, 0.0f);
  h[i] = v;
}

// one wave32 per node: log_softmax over C (<=64) classes, in place
__global__ void log_softmax_rows(float* __restrict__ h, int n, int C) {
  int node = blockIdx.x * 8 + (threadIdx.x >> 5);
  int lane = threadIdx.x & 31;
  if (node >= n) return;
  float* p = h + (size_t)node * C;
  float v0 = (lane < C) ? p[lane] : -INFINITY;
  float v1 = (lane + 32 < C) ? p[lane + 32] : -INFINITY;
  float m = wave_max(fmaxf(v0, v1));
  float s = 0.0f;
  if (lane < C) s += expf(v0 - m);
  if (lane + 32 < C) s += expf(v1 - m);
  s = wave_sum(s);
  float l = logf(s);
  if (lane < C) p[lane] = v0 - m - l;
  if (lane + 32 < C) p[lane + 32] = v1 - m - l;
}

// ---------------- host orchestration ----------------

static inline void run_att(const float* src, const int* row, const int* col,
                           float* xn, float* sim, float* rowsum, float* deg,
                           const float* adjbase, const float* gate, int gidx,
                           float* wout, int N, int E, hipStream_t s) {
  hipMemsetAsync(rowsum, 0, (size_t)N * 4, s);
  hipMemsetAsync(deg, 0, (size_t)N * 4, s);
  node_l2norm<<<(N + 7) / 8, 256, 0, s>>>(src, xn, N);
  edge_cos_sim<<<(E + 7) / 8, 256, 0, s>>>(xn, row, col, sim, E);
  edge_rowsum<<<(E + 255) / 256, 256, 0, s>>>(row, sim, rowsum, E);
  edge_norm_deg<<<(E + 255) / 256, 256, 0, s>>>(row, sim, rowsum, deg, E);
  edge_weight_final<<<(E + 255) / 256, 256, 0, s>>>(row, col, sim, deg, adjbase,
                                                    gate, gidx, wout, E);
}

static inline void run_conv(const float* hin, const int* row, const int* col,
                            const float* ew, const float* Wm, const float* bias,
                            float* hlin, float* deg2, float* dinv, float* hout,
                            int N, int E, int nOut, int doRelu, hipStream_t s) {
  int tilesN = (nOut + 15) / 16;
  int tiles = ((N + 15) / 16) * tilesN;
  gemm_wmma_f32<<<(tiles + 7) / 8, 256, 0, s>>>(hin, Wm, hlin, N, nOut, tilesN);
  hipMemsetAsync(deg2, 0, (size_t)N * 4, s);
  edge_colsum<<<(E + 255) / 256, 256, 0, s>>>(col, ew, deg2, E);
  node_rsqrt<<<(N + 255) / 256, 256, 0, s>>>(deg2, dinv, N);
  hipMemsetAsync(hout, 0, (size_t)N * nOut * 4, s);
  edge_aggregate<<<(E + 7) / 8, 256, 0, s>>>(row, col, ew, dinv, hlin, hout, E, nOut);
  long total = (long)N * nOut;
  bias_act<<<(int)((total + 255) / 256), 256, 0, s>>>(hout, bias, total, nOut, doRelu);
}

extern "C" void kernel_launch(void* const* d_in, const int* in_sizes, int n_in,
                              void* d_out, int out_size, void* d_ws, size_t ws_size,
                              hipStream_t stream) {
  const float* x    = (const float*)d_in[0];
  const float* gate = (const float*)d_in[1];
  const float* W0   = (const float*)d_in[2];
  const float* b0   = (const float*)d_in[3];
  const float* W1   = (const float*)d_in[4];
  const float* b1   = (const float*)d_in[5];
  const float* W2   = (const float*)d_in[6];
  const float* b2   = (const float*)d_in[7];
  const int*  eidx  = (const int*)d_in[8];

  const int N  = in_sizes[0] / FEAT;       // 50000
  const int E  = in_sizes[8] / 2;          // 850000
  const int C2 = in_sizes[6] / FEAT;       // 40
  const int* row = eidx;
  const int* col = eidx + E;

  char* ws = (char*)d_ws;
  size_t off = 0;
  auto take = [&](size_t bytes) -> float* {
    float* p = (float*)(ws + off);
    off = (off + bytes + 255) & ~(size_t)255;
    return p;
  };
  float* xn_hlin = take((size_t)N * FEAT * 4);  // xn during att, hlin during conv
  float* hA      = take((size_t)N * FEAT * 4);
  float* hB      = take((size_t)N * FEAT * 4);
  float* sim     = take((size_t)E * 4);
  float* adjbase = take((size_t)E * 4);
  float* wcur    = take((size_t)E * 4);
  float* rowsum  = take((size_t)N * 4);
  float* deg     = take((size_t)N * 4);
  float* deg2    = take((size_t)N * 4);
  float* dinv    = take((size_t)N * 4);
  (void)ws_size; (void)n_in; (void)out_size;

  // layer 0: base attention adjacency from raw features, then conv+relu
  run_att(x, row, col, xn_hlin, sim, rowsum, deg, nullptr, nullptr, -1, adjbase, N, E, stream);
  run_conv(x, row, col, adjbase, W0, b0, xn_hlin, deg2, dinv, hA, N, E, FEAT, 1, stream);

  // layer 1: new attention from h0, blend with gate[0], conv+relu
  run_att(hA, row, col, xn_hlin, sim, rowsum, deg, adjbase, gate, 0, wcur, N, E, stream);
  run_conv(hA, row, col, wcur, W1, b1, xn_hlin, deg2, dinv, hB, N, E, FEAT, 1, stream);

  // layer 2: blend with gate[1], conv (128 -> 40) into d_out, log_softmax
  run_att(hB, row, col, xn_hlin, sim, rowsum, deg, adjbase, gate, 1, wcur, N, E, stream);
  float* out = (float*)d_out;
  run_conv(hB, row, col, wcur, W2, b2, xn_hlin, deg2, dinv, out, N, E, C2, 0, stream);
  log_softmax_rows<<<(N + 7) / 8, 256, 0, stream>>>(out, N, C2);
}